// RNNDecoder_7146825580792
// MI455X (gfx1250) — compile-verified
//
#include <hip/hip_runtime.h>
#include <hip/hip_bf16.h>

typedef __attribute__((ext_vector_type(2))) float v2f;
typedef __attribute__((ext_vector_type(8))) float v8f;

#define B_SZ   32
#define T_SZ   64
#define D_SZ   1024
#define H_SZ   1024
#define V_SZ   32000
#define MXLEN_ 100
#define GO_TOK 1

// ---------------------------------------------------------------------------
// Generic f32 WMMA GEMM:
//   C[m,n] = init + sum_k A[rowmap(m), k] * B[k, n]
//   init = Cin[m,n] (if Cin) else bias[n] (if bias) else 0
// Block = 128 threads (4 wave32). WG tile = 16(M) x 256(N); wave tile 16x64.
// A tile (16x32) staged in LDS and shared by the 4 waves.
// Requires: M%16==0, N%256==0, K%32==0 (all call sites satisfy this).
// ---------------------------------------------------------------------------
__global__ __launch_bounds__(128)
void wmma_gemm_f32(const float* __restrict__ A,
                   const int*   __restrict__ rowmap,
                   const float* __restrict__ Bm,
                   const float* __restrict__ bias,
                   const float* __restrict__ Cin,
                   float*       __restrict__ C,
                   int M, int N, int K)
{
    __shared__ float As[16 * 33];           // 16x32 tile, padded to kill bank conflicts

    const int tid  = threadIdx.x;
    const int lane = tid & 31;
    const int wv   = tid >> 5;              // wave id 0..3
    const int kl   = lane >> 4;             // lane-half: 0 or 1
    const int ln   = lane & 15;             // 0..15
    const int m0   = blockIdx.y << 4;       // M tile base
    const int n0   = (blockIdx.x << 8) + (wv << 6);   // wave N base (64 cols)

    v8f acc[4];
    for (int j = 0; j < 4; ++j) {
        const int col = n0 + (j << 4) + ln;
        if (Cin) {
            for (int r = 0; r < 8; ++r)
                acc[j][r] = Cin[(size_t)(m0 + r + (kl << 3)) * N + col];
        } else if (bias) {
            const float bv = bias[col];
            for (int r = 0; r < 8; ++r) acc[j][r] = bv;
        } else {
            for (int r = 0; r < 8; ++r) acc[j][r] = 0.0f;
        }
    }

    for (int kb = 0; kb < K; kb += 32) {
        __syncthreads();
        // cooperatively stage A[16 x 32] (with optional row gather) into LDS
        for (int i = tid; i < 16 * 32; i += 128) {
            const int r  = i >> 5;
            const int cc = i & 31;
            const int ar = rowmap ? rowmap[m0 + r] : (m0 + r);
            As[r * 33 + cc] = A[(size_t)ar * K + kb + cc];
        }
        __syncthreads();

        if (kb + 32 < K) {   // hint next B block into cache (global_prefetch_b8)
            __builtin_prefetch(&Bm[(size_t)(kb + 32 + (kl << 1)) * N + n0 + ln], 0, 0);
        }

        for (int kk = 0; kk < 32; kk += 4) {
            // A fragment: 16x4 f32 -> 2 VGPRs/lane; lane L holds (M=L%16, K=2*(L/16)+v)
            v2f a;
            a.x = As[ln * 33 + kk + (kl << 1) + 0];
            a.y = As[ln * 33 + kk + (kl << 1) + 1];
            for (int j = 0; j < 4; ++j) {
                const int col = n0 + (j << 4) + ln;
                // B fragment: 4x16 f32 -> 2 VGPRs/lane; (K=2*(L/16)+v, N=L%16)
                v2f bf;
                bf.x = Bm[(size_t)(kb + kk + (kl << 1) + 0) * N + col];
                bf.y = Bm[(size_t)(kb + kk + (kl << 1) + 1) * N + col];
                acc[j] = __builtin_amdgcn_wmma_f32_16x16x4_f32(
                    /*neg_a=*/false, a, /*neg_b=*/false, bf,
                    /*c_mod=*/(short)0, acc[j],
                    /*reuse_a=*/false, /*reuse_b=*/false);
            }
        }
    }

    // C/D layout: VGPR r holds (M = r + 8*(lane/16), N = lane%16) within tile
    for (int j = 0; j < 4; ++j) {
        const int col = n0 + (j << 4) + ln;
        for (int r = 0; r < 8; ++r)
            C[(size_t)(m0 + r + (kl << 3)) * N + col] = acc[j][r];
    }
}

// ---------------------------------------------------------------------------
// LSTM gate nonlinearity: z[B,4H] (i,f,g,o) -> updates h,c; optional hs tap.
// c = sig(f+1)*c + sig(i)*tanh(g); h = sig(o)*tanh(c)   (TF forget_bias=1.0)
// ---------------------------------------------------------------------------
__global__ __launch_bounds__(256)
void lstm_gate(const float* __restrict__ z, float* __restrict__ h,
               float* __restrict__ c, float* __restrict__ hs_out)
{
    const int idx = blockIdx.x * blockDim.x + threadIdx.x;
    if (idx >= B_SZ * H_SZ) return;
    const int b = idx >> 10;            // H = 1024
    const int j = idx & (H_SZ - 1);
    const float* zr = z + (size_t)b * 4 * H_SZ;
    const float iv = zr[j];
    const float fv = zr[H_SZ + j];
    const float gv = zr[2 * H_SZ + j];
    const float ov = zr[3 * H_SZ + j];
    const float sf = 1.0f / (1.0f + __expf(-(fv + 1.0f)));
    const float si = 1.0f / (1.0f + __expf(-iv));
    const float so = 1.0f / (1.0f + __expf(-ov));
    const float cn = sf * c[idx] + si * tanhf(gv);
    const float hn = so * tanhf(cn);
    c[idx] = cn;
    h[idx] = hn;
    if (hs_out) hs_out[idx] = hn;
}

// ---------------------------------------------------------------------------
// Row softmax over V columns (one workgroup per row).
// ---------------------------------------------------------------------------
__global__ __launch_bounds__(256)
void softmax_rows(const float* __restrict__ logits, float* __restrict__ probs, int N)
{
    __shared__ float red[256];
    const int row = blockIdx.x;
    const float* x = logits + (size_t)row * N;
    float*       y = probs  + (size_t)row * N;

    float mx = -3.402823466e+38f;
    for (int i = threadIdx.x; i < N; i += 256) mx = fmaxf(mx, x[i]);
    red[threadIdx.x] = mx; __syncthreads();
    for (int s = 128; s > 0; s >>= 1) {
        if (threadIdx.x < s) red[threadIdx.x] = fmaxf(red[threadIdx.x], red[threadIdx.x + s]);
        __syncthreads();
    }
    mx = red[0]; __syncthreads();

    float sum = 0.0f;
    for (int i = threadIdx.x; i < N; i += 256) {
        const float e = __expf(x[i] - mx);
        y[i] = e;
        sum += e;
    }
    red[threadIdx.x] = sum; __syncthreads();
    for (int s = 128; s > 0; s >>= 1) {
        if (threadIdx.x < s) red[threadIdx.x] += red[threadIdx.x + s];
        __syncthreads();
    }
    const float inv = 1.0f / red[0];
    for (int i = threadIdx.x; i < N; i += 256) y[i] *= inv;
}

// ---------------------------------------------------------------------------
// Row argmax over V columns (one workgroup per row); first-max tie-breaking.
// Writes next-token index (int) and the float-cast token into best_out.
// ---------------------------------------------------------------------------
__global__ __launch_bounds__(256)
void argmax_rows(const float* __restrict__ lg, int* __restrict__ tok,
                 float* __restrict__ best_out, int N)
{
    __shared__ float vred[256];
    __shared__ int   ired[256];
    const int row = blockIdx.x;
    const float* x = lg + (size_t)row * N;

    float bv = -3.402823466e+38f;
    int   bi = 0;
    for (int i = threadIdx.x; i < N; i += 256) {
        const float v = x[i];
        if (v > bv) { bv = v; bi = i; }
    }
    vred[threadIdx.x] = bv; ired[threadIdx.x] = bi; __syncthreads();
    for (int s = 128; s > 0; s >>= 1) {
        if (threadIdx.x < s) {
            const float ov = vred[threadIdx.x + s];
            const int   oi = ired[threadIdx.x + s];
            if (ov > vred[threadIdx.x] ||
                (ov == vred[threadIdx.x] && oi < ired[threadIdx.x])) {
                vred[threadIdx.x] = ov; ired[threadIdx.x] = oi;
            }
        }
        __syncthreads();
    }
    if (threadIdx.x == 0) {
        tok[row]      = ired[0];
        best_out[row] = (float)ired[0];
    }
}

// ---------------------------------------------------------------------------
// Small helpers
// ---------------------------------------------------------------------------
__global__ void zero_f32(float* p, int n)
{
    const int i = blockIdx.x * blockDim.x + threadIdx.x;
    if (i < n) p[i] = 0.0f;
}

__global__ void build_rowmap(const int* __restrict__ tgt, int* __restrict__ rowmap)
{
    const int m = blockIdx.x * blockDim.x + threadIdx.x;   // m = t*B + b
    if (m < T_SZ * B_SZ) {
        const int t = m / B_SZ;
        const int b = m % B_SZ;
        rowmap[m] = tgt[b * T_SZ + t];                      // tgt is [B,T]
    }
}

__global__ void init_tok(int* tok)
{
    const int i = threadIdx.x;
    if (i < B_SZ) tok[i] = GO_TOK;
}

// ---------------------------------------------------------------------------
extern "C" void kernel_launch(void* const* d_in, const int* in_sizes, int n_in,
                              void* d_out, int out_size, void* d_ws, size_t ws_size,
                              hipStream_t stream)
{
    const int*   tgt  = (const int*)  d_in[0];   // [B,T]
    const float* Wo   = (const float*)d_in[1];   // [V,D]
    const float* Wx   = (const float*)d_in[2];   // [D,4H]
    const float* Wh   = (const float*)d_in[3];   // [H,4H]
    const float* bias = (const float*)d_in[4];   // [4H]
    const float* Wp   = (const float*)d_in[5];   // [H,V]

    float* logits = (float*)d_out;                               // [T*B, V]
    float* probs  = logits + (size_t)T_SZ * B_SZ * V_SZ;         // [T*B, V]
    float* best   = probs  + (size_t)T_SZ * B_SZ * V_SZ;         // [MXLEN, B] as float

    char* wsp = (char*)d_ws;
    float* XWx = (float*)wsp; wsp += (size_t)T_SZ * B_SZ * 4 * H_SZ * sizeof(float); // 33.5 MB
    float* hs  = (float*)wsp; wsp += (size_t)T_SZ * B_SZ * H_SZ * sizeof(float);     //  8.4 MB
    float* z   = (float*)wsp; wsp += (size_t)B_SZ * 4 * H_SZ * sizeof(float);
    float* h   = (float*)wsp; wsp += (size_t)B_SZ * H_SZ * sizeof(float);
    float* c   = (float*)wsp; wsp += (size_t)B_SZ * H_SZ * sizeof(float);
    float* lg  = (float*)wsp; wsp += (size_t)B_SZ * V_SZ * sizeof(float);
    int* rowmap = (int*)wsp;  wsp += (size_t)T_SZ * B_SZ * sizeof(int);
    int* tok    = (int*)wsp;

    const dim3 gblk(128);
    const int  nBH = B_SZ * H_SZ;

    // ---- Phase 0: hoist the non-serial work -------------------------------
    build_rowmap<<<(T_SZ * B_SZ + 127) / 128, 128, 0, stream>>>(tgt, rowmap);
    // XWx[t*B+b, :] = bias + Wo[tgt[b,t]] @ Wx   (2048 x 4096 x 1024, one GEMM)
    wmma_gemm_f32<<<dim3(4 * H_SZ / 256, (T_SZ * B_SZ) / 16), gblk, 0, stream>>>(
        Wo, rowmap, Wx, bias, nullptr, XWx, T_SZ * B_SZ, 4 * H_SZ, D_SZ);

    zero_f32<<<(nBH + 255) / 256, 256, 0, stream>>>(h, nBH);
    zero_f32<<<(nBH + 255) / 256, 256, 0, stream>>>(c, nBH);

    // ---- Phase 1: teacher-forced recurrence (serial; only h@Wh per step) --
    for (int t = 0; t < T_SZ; ++t) {
        wmma_gemm_f32<<<dim3(4 * H_SZ / 256, B_SZ / 16), gblk, 0, stream>>>(
            h, nullptr, Wh, nullptr, XWx + (size_t)t * B_SZ * 4 * H_SZ, z,
            B_SZ, 4 * H_SZ, H_SZ);
        lstm_gate<<<(nBH + 255) / 256, 256, 0, stream>>>(
            z, h, c, hs + (size_t)t * B_SZ * H_SZ);
    }

    // ---- Phase 2: big projection + softmax (Wp lives in the 192MB L2) ----
    wmma_gemm_f32<<<dim3(V_SZ / 256, (T_SZ * B_SZ) / 16), gblk, 0, stream>>>(
        hs, nullptr, Wp, nullptr, nullptr, logits, T_SZ * B_SZ, V_SZ, H_SZ);
    softmax_rows<<<T_SZ * B_SZ, 256, 0, stream>>>(logits, probs, V_SZ);

    // ---- Phase 3: greedy decode (100 serial steps) ------------------------
    zero_f32<<<(nBH + 255) / 256, 256, 0, stream>>>(h, nBH);
    zero_f32<<<(nBH + 255) / 256, 256, 0, stream>>>(c, nBH);
    init_tok<<<1, 32, 0, stream>>>(tok);

    for (int s = 0; s < MXLEN_; ++s) {
        // z = bias + Wo[tok] @ Wx
        wmma_gemm_f32<<<dim3(4 * H_SZ / 256, B_SZ / 16), gblk, 0, stream>>>(
            Wo, tok, Wx, bias, nullptr, z, B_SZ, 4 * H_SZ, D_SZ);
        // z += h @ Wh
        wmma_gemm_f32<<<dim3(4 * H_SZ / 256, B_SZ / 16), gblk, 0, stream>>>(
            h, nullptr, Wh, nullptr, z, z, B_SZ, 4 * H_SZ, H_SZ);
        lstm_gate<<<(nBH + 255) / 256, 256, 0, stream>>>(z, h, c, nullptr);
        // lg = h @ Wp
        wmma_gemm_f32<<<dim3(V_SZ / 256, B_SZ / 16), gblk, 0, stream>>>(
            h, nullptr, Wp, nullptr, nullptr, lg, B_SZ, V_SZ, H_SZ);
        argmax_rows<<<B_SZ, 256, 0, stream>>>(lg, tok, best + (size_t)s * B_SZ, V_SZ);
    }
}